// HouseholderReflection_59433757442157
// MI455X (gfx1250) — compile-verified
//
#include <hip/hip_runtime.h>

// ---------------------------------------------------------------------------
// Householder scan  ->  Z = X * Q,  Q = H_1 H_2 ... H_R   (H_r = I - 2 vr^ vr^T)
//
// Kernel 1 (1 workgroup, 8 waves): blocked compact-WY composition of Q in the
//   320KB WGP LDS; all heavy FLOPs on V_WMMA_F32_16X16X4_F32.
// Kernel 2: Z = X*Q via WMMA, K-panels double-buffered in LDS with
//   GLOBAL_LOAD_ASYNC_TO_LDS_B128 (ASYNCcnt) when the toolchain exposes it.
// ---------------------------------------------------------------------------

typedef float v2f __attribute__((ext_vector_type(2)));
typedef float v8f __attribute__((ext_vector_type(8)));

#define DD 256
#define RR 256

// ---------------- kernel 1 LDS layout (floats) ----------------
#define QST 260
#define VST 260
#define WST 20
#define SST 20
#define QS_OFF  0
#define VB_OFF  (QS_OFF + DD * QST)
#define WS_OFF  (VB_OFF + 16 * VST)
#define SNT_OFF (WS_OFF + DD * WST)
#define GS_OFF  (SNT_OFF + 16 * SST)
#define SS_OFF  (GS_OFF + 16 * 17)
#define INV_OFF (SS_OFF + 16 * 17)
#define K1_FLOATS (INV_OFF + 256)           // 307840 B < 320KB

__global__ __launch_bounds__(256) void hh_compose_qt(
    const float* __restrict__ V, float* __restrict__ QT)
{
    extern __shared__ float smem[];
    float* Qs  = smem + QS_OFF;
    float* Vb  = smem + VB_OFF;
    float* Ws  = smem + WS_OFF;
    float* SnT = smem + SNT_OFF;
    float* Gs  = smem + GS_OFF;
    float* Ss  = smem + SS_OFF;
    float* inv_norm = smem + INV_OFF;

    const int tid   = threadIdx.x;
    const int wave  = tid >> 5;
    const int lane  = tid & 31;
    const int lrow  = lane & 15;
    const int hi    = lane >> 4;
    const int khalf = hi << 1;

    {
        float ss = 0.f;
        const float4* vr = (const float4*)(V + (size_t)tid * DD);
        #pragma unroll 4
        for (int i = 0; i < DD / 4; ++i) {
            float4 p = vr[i];
            ss += p.x * p.x + p.y * p.y + p.z * p.z + p.w * p.w;
        }
        inv_norm[tid] = 1.0f / (sqrtf(ss) + 1e-8f);
    }
    for (int c = 0; c < DD; ++c)
        Qs[tid * QST + c] = (tid == c) ? 1.0f : 0.0f;
    __syncthreads();

    const v8f vzero = (v8f){0.f,0.f,0.f,0.f,0.f,0.f,0.f,0.f};

    for (int b = 0; b < RR / 16; ++b) {
        #pragma unroll
        for (int i = 0; i < 16; ++i) {
            int idx = tid + i * 256;
            int r = idx >> 8, c = idx & 255;
            Vb[r * VST + c] = V[(size_t)(b * 16 + r) * DD + c] * inv_norm[b * 16 + r];
        }
        __syncthreads();

        if (wave == 0) {                       // G = V_b V_b^T
            v8f g = vzero;
            for (int k0 = 0; k0 < DD; k0 += 4) {
                v2f a = *(const v2f*)&Vb[lrow * VST + k0 + khalf];
                g = __builtin_amdgcn_wmma_f32_16x16x4_f32(
                        false, a, false, a, (short)0, g, false, false);
            }
            #pragma unroll
            for (int j = 0; j < 8; ++j)
                Gs[(j + 8 * hi) * 17 + lrow] = g[j];
        }
        __syncthreads();

        if (wave == 0 && lane < 16) {          // triangular recurrence for S
            const int i = lane;
            for (int k = 0; k < 16; ++k) {
                float s = 0.f;
                if (i == k) s = 2.f;
                else if (i < k) {
                    float acc = 0.f;
                    for (int j = i; j < k; ++j)
                        acc += Ss[i * 17 + j] * Gs[j * 17 + k];
                    s = -2.f * acc;
                }
                Ss[i * 17 + k] = s;
            }
            for (int k = 0; k < 16; ++k)
                SnT[i * SST + k] = -Ss[k * 17 + i];
        }
        __syncthreads();

        #pragma unroll
        for (int t = 0; t < 2; ++t) {
            const int mt = 2 * wave + t;
            const int mrow = mt * 16;

            v8f w = vzero;                     // W = Q * V_b^T
            for (int k0 = 0; k0 < DD; k0 += 4) {
                v2f a = *(const v2f*)&Qs[(mrow + lrow) * QST + k0 + khalf];
                v2f bb = *(const v2f*)&Vb[lrow * VST + k0 + khalf];
                w = __builtin_amdgcn_wmma_f32_16x16x4_f32(
                        false, a, false, bb, (short)0, w, false, false);
            }
            #pragma unroll
            for (int j = 0; j < 8; ++j)
                Ws[(mrow + j + 8 * hi) * WST + lrow] = w[j];

            v8f y = vzero;                     // Y = W * (-S)
            #pragma unroll
            for (int k0 = 0; k0 < 16; k0 += 4) {
                v2f a = *(const v2f*)&Ws[(mrow + lrow) * WST + k0 + khalf];
                v2f bb = *(const v2f*)&SnT[lrow * SST + k0 + khalf];
                y = __builtin_amdgcn_wmma_f32_16x16x4_f32(
                        false, a, false, bb, (short)0, y, false, false);
            }
            #pragma unroll
            for (int j = 0; j < 8; ++j)
                Ws[(mrow + j + 8 * hi) * WST + lrow] = y[j];

            for (int nt = 0; nt < 16; ++nt) {  // Q += Y * V_b  (C = Q tile)
                v8f c;
                #pragma unroll
                for (int j = 0; j < 8; ++j)
                    c[j] = Qs[(mrow + j + 8 * hi) * QST + nt * 16 + lrow];
                #pragma unroll
                for (int k0 = 0; k0 < 16; k0 += 4) {
                    const int kk = k0 + khalf;
                    v2f a = *(const v2f*)&Ws[(mrow + lrow) * WST + kk];
                    v2f bb;
                    bb.x = Vb[kk * VST + nt * 16 + lrow];
                    bb.y = Vb[(kk + 1) * VST + nt * 16 + lrow];
                    c = __builtin_amdgcn_wmma_f32_16x16x4_f32(
                            false, a, false, bb, (short)0, c, false, false);
                }
                #pragma unroll
                for (int j = 0; j < 8; ++j)
                    Qs[(mrow + j + 8 * hi) * QST + nt * 16 + lrow] = c[j];
            }
        }
        __syncthreads();
    }

    for (int row = 0; row < DD; ++row)
        QT[(size_t)row * DD + tid] = Qs[tid * QST + row];
}

// ---------------------------------------------------------------------------
// Kernel 2: Z = X * Q with async double-buffered LDS staging when available.
// ---------------------------------------------------------------------------
#define BM 128
#define BK 64
#define XS_STRIDE 76
#define QT_STRIDE 76
#define PANEL_FLOATS (BM * XS_STRIDE + DD * QT_STRIDE)   // 29184 fl = 116736 B

#if defined(__HIP_DEVICE_COMPILE__) && defined(__has_builtin)
#  if __has_builtin(__builtin_amdgcn_global_load_async_to_lds_b128) && \
      __has_builtin(__builtin_amdgcn_s_wait_asynccnt)
#    define HAVE_ASYNC_LDS 1
#  endif
#endif
#ifndef HAVE_ASYNC_LDS
#  define HAVE_ASYNC_LDS 0
#endif

#if HAVE_ASYNC_LDS
typedef int v4i_gcc __attribute__((vector_size(4 * sizeof(int))));
__device__ __forceinline__ void async_cp16(const float* g, float* l) {
    // b128 global -> LDS, tracked by ASYNCcnt (no VGPR round-trip).
    // Toolchain signature (from diagnostic): generic `int __vector(4)*` params.
    __builtin_amdgcn_global_load_async_to_lds_b128(
        (v4i_gcc*)g, (v4i_gcc*)l, 0, 0);
}
#endif

__device__ __forceinline__ void panel_mm(const float* Xs, const float* Qts,
                                         v8f* acc, int wave, int lrow, int khalf)
{
    for (int k0 = 0; k0 < BK; k0 += 4) {
        const int kk = k0 + khalf;
        v2f a = *(const v2f*)&Xs[(wave * 16 + lrow) * XS_STRIDE + kk];
        #pragma unroll
        for (int nt = 0; nt < 16; ++nt) {
            v2f b = *(const v2f*)&Qts[(nt * 16 + lrow) * QT_STRIDE + kk];
            acc[nt] = __builtin_amdgcn_wmma_f32_16x16x4_f32(
                false, a, false, b, (short)0, acc[nt], false, false);
        }
    }
}

__global__ __launch_bounds__(256) void hh_apply_q_wmma(
    const float* __restrict__ X, const float* __restrict__ QT,
    float* __restrict__ Z, float* __restrict__ logdet)
{
    extern __shared__ float smem[];
    const int tid   = threadIdx.x;
    const int wave  = tid >> 5;
    const int lane  = tid & 31;
    const int lrow  = lane & 15;
    const int khalf = (lane >> 4) << 1;
    const long long blockRow0 = (long long)blockIdx.x * BM;

    v8f acc[16];
    #pragma unroll
    for (int i = 0; i < 16; ++i) acc[i] = (v8f){0.f,0.f,0.f,0.f,0.f,0.f,0.f,0.f};

#if HAVE_ASYNC_LDS
    auto issue = [&](int kp, int pb) {
        float* Xs  = smem + pb * PANEL_FLOATS;
        float* Qts = Xs + BM * XS_STRIDE;
        #pragma unroll
        for (int i = 0; i < 8; ++i) {
            int idx = tid + i * 256;
            int rr = idx >> 4, k4 = (idx & 15) << 2;
            async_cp16(&X[(blockRow0 + rr) * DD + kp + k4], &Xs[rr * XS_STRIDE + k4]);
        }
        #pragma unroll
        for (int i = 0; i < 16; ++i) {
            int idx = tid + i * 256;
            int nn = idx >> 4, k4 = (idx & 15) << 2;
            async_cp16(&QT[(size_t)nn * DD + kp + k4], &Qts[nn * QT_STRIDE + k4]);
        }
    };

    issue(0, 0);
    for (int p = 0; p < DD / BK; ++p) {
        __builtin_amdgcn_s_wait_asynccnt(0);  // my copies done
        __syncthreads();                      // everyone's copies done; prev buf free
        if (p + 1 < DD / BK) issue((p + 1) * BK, (p + 1) & 1);
        const float* Xs  = smem + (p & 1) * PANEL_FLOATS;
        const float* Qts = Xs + BM * XS_STRIDE;
        panel_mm(Xs, Qts, acc, wave, lrow, khalf);
    }
#else
    float* Xs  = smem;
    float* Qts = smem + BM * XS_STRIDE;
    for (int kp = 0; kp < DD; kp += BK) {
        #pragma unroll
        for (int i = 0; i < 8; ++i) {
            int idx = tid + i * 256;
            int rr = idx >> 4, k4 = (idx & 15) << 2;
            *(float4*)&Xs[rr * XS_STRIDE + k4] =
                *(const float4*)&X[(blockRow0 + rr) * DD + kp + k4];
        }
        #pragma unroll
        for (int i = 0; i < 16; ++i) {
            int idx = tid + i * 256;
            int nn = idx >> 4, k4 = (idx & 15) << 2;
            *(float4*)&Qts[nn * QT_STRIDE + k4] =
                *(const float4*)&QT[(size_t)nn * DD + kp + k4];
        }
        __syncthreads();
        panel_mm(Xs, Qts, acc, wave, lrow, khalf);
        __syncthreads();
    }
#endif

    const int rbase = wave * 16 + 8 * (lane >> 4);
    #pragma unroll
    for (int nt = 0; nt < 16; ++nt) {
        const int col = nt * 16 + lrow;
        #pragma unroll
        for (int j = 0; j < 8; ++j)
            Z[(blockRow0 + rbase + j) * DD + col] = acc[nt][j];
    }

    if (tid < BM) logdet[blockRow0 + tid] = 0.0f;
}

// ---------------------------------------------------------------------------
extern "C" void kernel_launch(void* const* d_in, const int* in_sizes, int n_in,
                              void* d_out, int out_size, void* d_ws, size_t ws_size,
                              hipStream_t stream) {
    const float* X = (const float*)d_in[0];
    const float* V = (const float*)d_in[1];
    const long long BD = in_sizes[0];
    const long long B  = BD / DD;

    float* out    = (float*)d_out;
    float* Z      = out;
    float* logdet = out + BD;
    float* QT     = (float*)d_ws;

    const int lds1 = K1_FLOATS * (int)sizeof(float);
    const int lds2 = 2 * PANEL_FLOATS * (int)sizeof(float);  // always double-buffer size

    (void)hipFuncSetAttribute((const void*)hh_compose_qt,
                              hipFuncAttributeMaxDynamicSharedMemorySize, lds1);
    (void)hipFuncSetAttribute((const void*)hh_apply_q_wmma,
                              hipFuncAttributeMaxDynamicSharedMemorySize, lds2);

    hh_compose_qt<<<1, 256, lds1, stream>>>(V, QT);

    const int grid = (int)(B / BM);
    hh_apply_q_wmma<<<grid, 256, lds2, stream>>>(X, QT, Z, logdet);
}